// GAT_89876485636806
// MI455X (gfx1250) — compile-verified
//
#include <hip/hip_runtime.h>
#include <hip/hip_bf16.h>

#define NNODE 4096
#define HWDIM 512
#define CCH   64
#define HIDD  8
#define LRELU 0.2f
#define LOG2E 1.4426950408889634f

#define TPAD   36            // padded row stride (floats): 36*4=144 B keeps 16B alignment,
                             // and (r*36 + c) mod 64 banks are conflict-free for the A reads
#define TILE_F (16 * TPAD)   // floats per adj tile buffer (16 rows x 32 cols, padded)

typedef __attribute__((ext_vector_type(2))) float v2f;
typedef __attribute__((ext_vector_type(8))) float v8f;

// ---------------- workspace layout (float offsets) ----------------
enum : int {
  WS_FTS1  = 0,                       // 4096*8
  WS_F1    = WS_FTS1 + NNODE * HIDD,
  WS_F2    = WS_F1 + NNODE,
  WS_H     = WS_F2 + NNODE,           // 4096*8
  WS_FTS2  = WS_H + NNODE * HIDD,
  WS_F1B   = WS_FTS2 + NNODE,
  WS_F2B   = WS_F1B + NNODE,
  WS_BPACK = WS_F2B + NNODE,          // (4096/4)*128 = 131072 (B0,B1,f2a,f2b per lane)
  WS_MAX0  = WS_BPACK + (NNODE / 4) * 128,
  WS_MAX1  = WS_MAX0 + 1,
};

// ---------------- output layout (node_logits, h_1, node_byxs, node_labels) ----------------
enum : int {
  OUT_LOGITS = 0,
  OUT_H1  = NNODE,                    // 4096*64
  OUT_BYX = OUT_H1 + NNODE * CCH,     // 8192
  OUT_LBL = OUT_BYX + NNODE * 2,      // 4096
};

// Gather node features, project x@W1.T -> fts (N x 8), compute f1/f2 scalars,
// emit node_byxs (as float) and node_labels outputs.
__global__ void k_prep1(const int* __restrict__ nb, const float* __restrict__ conv,
                        const float* __restrict__ labels, const float* __restrict__ W1,
                        const float* __restrict__ a1w, const float* __restrict__ a1b,
                        const float* __restrict__ a2w, const float* __restrict__ a2b,
                        float* __restrict__ ws, float* __restrict__ out) {
  int i = blockIdx.x * blockDim.x + threadIdx.x;
  if (i >= NNODE) return;
  int by = nb[2 * i], bx = nb[2 * i + 1];
  const float4* xp = (const float4*)(conv + ((size_t)by * HWDIM + bx) * CCH);
  float fts[HIDD];
#pragma unroll
  for (int r = 0; r < HIDD; ++r) fts[r] = 0.f;
#pragma unroll
  for (int c4 = 0; c4 < CCH / 4; ++c4) {
    float4 xv = xp[c4];
#pragma unroll
    for (int r = 0; r < HIDD; ++r) {
      const float4 wv = *(const float4*)(W1 + r * CCH + c4 * 4);  // uniform -> s_load
      fts[r] = fmaf(xv.x, wv.x, fmaf(xv.y, wv.y, fmaf(xv.z, wv.z, fmaf(xv.w, wv.w, fts[r]))));
    }
  }
  float f1 = a1b[0], f2 = a2b[0];
#pragma unroll
  for (int r = 0; r < HIDD; ++r) {
    f1 = fmaf(a1w[r], fts[r], f1);
    f2 = fmaf(a2w[r], fts[r], f2);
  }
#pragma unroll
  for (int r = 0; r < HIDD; ++r) ws[WS_FTS1 + i * HIDD + r] = fts[r];
  ws[WS_F1 + i] = f1;
  ws[WS_F2 + i] = f2;
  out[OUT_BYX + 2 * i]     = (float)by;
  out[OUT_BYX + 2 * i + 1] = (float)bx;
  out[OUT_LBL + i] = labels[(size_t)by * HWDIM + bx];
}

// Max-reduce n floats -> *out (single block).
__global__ void k_rmax(const float* __restrict__ in, int n, float* __restrict__ out) {
  __shared__ float sm[256];
  int t = threadIdx.x;
  float m = -3.4e38f;
  for (int i = t; i < n; i += 256) m = fmaxf(m, in[i]);
  sm[t] = m;
  __syncthreads();
  for (int s = 128; s > 0; s >>= 1) {
    if (t < s) sm[t] = fmaxf(sm[t], sm[t + s]);
    __syncthreads();
  }
  if (!t) *out = sm[0];
}

// Pack per (K-group g, lane) a float4: {B_vgpr0, B_vgpr1, f2[j_k0], f2[j_k0+1]}.
// B is the 4x16 (KxN) WMMA operand: VGPR0 holds K={0|2} per lane-half, VGPR1 K={1|3}.
// Column `onescol` of B is all ones so the WMMA also accumulates the softmax denom Z.
__global__ void k_bpack(const float* __restrict__ fts, const float* __restrict__ f2arr,
                        int ncols, int onescol, float* __restrict__ bp) {
  int idx = blockIdx.x * blockDim.x + threadIdx.x;  // (N/4)*128 total
  int g = idx >> 7;
  int slot = idx & 127;
  int lane = slot >> 2;
  int q = slot & 3;
  int c = lane & 15;
  int k0 = (lane & 16) ? 2 : 0;
  float v;
  if (q < 2) {                      // B operand element, K = k0 + q
    int j = g * 4 + k0 + q;
    v = (c < ncols) ? fts[j * ncols + c] : (c == onescol ? 1.f : 0.f);
  } else {                          // f2 pair for this lane's A columns
    int j = g * 4 + k0 + (q - 2);
    v = f2arr[j];
  }
  bp[idx] = v;
}

// Fused dense GAT attention row-tile kernel. One block = 16 rows, 4 waves split the
// 4096 columns. Each wave double-buffers 16x32 adj tiles into LDS with
// global_load_async_to_lds_b128 (fully coalesced: 8 lanes cover a 128B row segment),
// then per K=4 chunk reads the A pair from LDS (conflict-free ds_load_b64), fetches
// the packed B/f2 float4, computes w = exp(lrelu(adj*(f1+f2)) - M) into the WMMA A
// layout, and a single V_WMMA_F32_16X16X4_F32 accumulates numerator + Z.
template <int LAYER>
__launch_bounds__(128) __global__
void k_attn(const float* __restrict__ adj, const float* __restrict__ f1arr,
            const float* __restrict__ bpack, const float* __restrict__ maxp,
            const float* __restrict__ bias, float* __restrict__ hws,
            float* __restrict__ out) {
  __shared__ float tile[4][2][TILE_F];  // per-wave double-buffered adj tiles (18 KB)
  __shared__ float red[4 * 8 * 32];     // cross-wave D reduction (4 KB)

  const int tid  = threadIdx.x;
  const int wave = tid >> 5;
  const int lane = tid & 31;
  const int rowbase = blockIdx.x * 16;
  const int k0 = (lane & 16) ? 2 : 0;   // ISA A-layout: lanes 0-15 K={0,1}, 16-31 K={2,3}
  const float F2max = maxp[0];
  const float f1v = f1arr[rowbase + (lane & 15)];
  const float Mi = fmaxf(0.f, f1v + F2max);   // upper bound on row's lrelu'd logits

  const int colbase = wave * (NNODE / 4);     // 1024 columns per wave, 32 chunks of 32
  const int rowl = lane >> 3;                 // 0..3: which of 4 rows this lane stages
  const int colq = (lane & 7) * 4;            // 0..28: 4-float column slot within chunk

  // Per-lane global byte offsets (chunk 0) and LDS byte addresses (buffer 0) for the
  // 4 staging instructions: instruction i covers rows 4i..4i+3, 32 columns.
  unsigned goffb[4], ldsb[4];
#pragma unroll
  for (int i = 0; i < 4; ++i) {
    goffb[i] = (unsigned)((rowbase + 4 * i + rowl) * (NNODE * 4) + (colbase + colq) * 4);
    ldsb[i]  = (unsigned)(size_t)&tile[wave][0][(4 * i + rowl) * TPAD + colq];
  }

  auto stage = [&](int ch, int buf) {
    // WAR guard: make sure our pending LDS reads of this buffer are drained.
    asm volatile("s_wait_dscnt 0" ::: "memory");
    const unsigned cb = (unsigned)(ch * 128);          // 32 cols * 4B per chunk
    const unsigned bo = (unsigned)(buf * (TILE_F * 4));
#pragma unroll
    for (int i = 0; i < 4; ++i) {
      unsigned go = goffb[i] + cb;
      unsigned ld = ldsb[i] + bo;
      asm volatile("global_load_async_to_lds_b128 %0, %1, %2"
                   :
                   : "v"(ld), "v"(go), "s"(adj)
                   : "memory");
    }
  };

  v8f acc = {0.f, 0.f, 0.f, 0.f, 0.f, 0.f, 0.f, 0.f};
  stage(0, 0);
  for (int ch = 0; ch < 32; ++ch) {
    const int buf = ch & 1;
    if (ch + 1 < 32) {
      stage(ch + 1, (ch + 1) & 1);
      asm volatile("s_wait_asynccnt 4" ::: "memory");  // current chunk's 4 loads done
    } else {
      asm volatile("s_wait_asynccnt 0" ::: "memory");
    }
    const int gbase = (colbase >> 2) + ch * 8;
#pragma unroll
    for (int kk = 0; kk < 8; ++kk) {
      const float4 pk = *(const float4*)(bpack + (size_t)(gbase + kk) * 128 + lane * 4);
      const float2 ap = *(const float2*)&tile[wave][buf][(lane & 15) * TPAD + kk * 4 + k0];
      v2f a, b;
      float t0 = ap.x * (f1v + pk.z);
      float s0 = t0 > 0.f ? t0 : LRELU * t0;
      a[0] = __builtin_amdgcn_exp2f((s0 - Mi) * LOG2E);
      float t1 = ap.y * (f1v + pk.w);
      float s1 = t1 > 0.f ? t1 : LRELU * t1;
      a[1] = __builtin_amdgcn_exp2f((s1 - Mi) * LOG2E);
      b[0] = pk.x;
      b[1] = pk.y;
      acc = __builtin_amdgcn_wmma_f32_16x16x4_f32(false, a, false, b, (short)0, acc,
                                                  false, false);
    }
  }

  // Cross-wave partial-sum reduction, then finalize in wave 0.
#pragma unroll
  for (int v = 0; v < 8; ++v) red[wave * 256 + v * 32 + lane] = acc[v];
  __syncthreads();
  if (tid < 32) {
#pragma unroll
    for (int v = 0; v < 8; ++v) {
      float sum = red[v * 32 + lane] + red[256 + v * 32 + lane] +
                  red[512 + v * 32 + lane] + red[768 + v * 32 + lane];
      const int c = lane & 15;
      const int row = rowbase + v + ((lane & 16) ? 8 : 0);  // D layout: M = v (+8 hi half)
      if (LAYER == 1) {
        float Z = __shfl(sum, (lane & 16) + 8, 32);   // Z lives in column 8
        float o = sum / Z + bias[c < 8 ? c : 0];
        float hval = o > 0.f ? o : (__builtin_amdgcn_exp2f(o * LOG2E) - 1.f);  // ELU
        if (c < 8) {
          hws[row * HIDD + c] = hval;
#pragma unroll
          for (int r = 0; r < 8; ++r)  // h_1 = concat([h]*8)
            out[OUT_H1 + row * CCH + r * HIDD + c] = hval;
        }
      } else {
        float Z = __shfl(sum, (lane & 16) + 1, 32);   // Z lives in column 1
        if (c == 0) out[OUT_LOGITS + row] = sum / Z + bias[0];
      }
    }
  }
}

// Layer-2 projection: fts2 = h_1 @ Wo.T collapses to h @ wo_eff (wo_eff[c] = sum_r Wo[r*8+c]).
__global__ void k_prep2(const float* __restrict__ hws, const float* __restrict__ Wo,
                        const float* __restrict__ o1w, const float* __restrict__ o1b,
                        const float* __restrict__ o2w, const float* __restrict__ o2b,
                        float* __restrict__ ws) {
  int i = blockIdx.x * blockDim.x + threadIdx.x;
  if (i >= NNODE) return;
  float f = 0.f;
#pragma unroll
  for (int c = 0; c < 8; ++c) {
    float wsum = 0.f;
#pragma unroll
    for (int r = 0; r < 8; ++r) wsum += Wo[r * 8 + c];  // uniform -> s_load
    f = fmaf(hws[i * 8 + c], wsum, f);
  }
  ws[WS_FTS2 + i] = f;
  ws[WS_F1B + i] = fmaf(f, o1w[0], o1b[0]);
  ws[WS_F2B + i] = fmaf(f, o2w[0], o2b[0]);
}

extern "C" void kernel_launch(void* const* d_in, const int* in_sizes, int n_in,
                              void* d_out, int out_size, void* d_ws, size_t ws_size,
                              hipStream_t stream) {
  const int*   nb     = (const int*)  d_in[0];
  const float* adj    = (const float*)d_in[1];
  const float* conv   = (const float*)d_in[2];
  const float* labels = (const float*)d_in[3];
  const float* W1     = (const float*)d_in[4];
  const float* a1w    = (const float*)d_in[5];
  const float* a1b    = (const float*)d_in[6];
  const float* a2w    = (const float*)d_in[7];
  const float* a2b    = (const float*)d_in[8];
  const float* bias1  = (const float*)d_in[9];
  const float* Wo     = (const float*)d_in[10];
  const float* o1w    = (const float*)d_in[11];
  const float* o1b    = (const float*)d_in[12];
  const float* o2w    = (const float*)d_in[13];
  const float* o2b    = (const float*)d_in[14];
  const float* biaso  = (const float*)d_in[15];
  float* out = (float*)d_out;
  float* ws  = (float*)d_ws;

  // Layer 1
  k_prep1<<<NNODE / 128, 128, 0, stream>>>(nb, conv, labels, W1, a1w, a1b, a2w, a2b, ws, out);
  k_rmax<<<1, 256, 0, stream>>>(ws + WS_F2, NNODE, ws + WS_MAX0);
  k_bpack<<<(NNODE / 4) * 128 / 256, 256, 0, stream>>>(ws + WS_FTS1, ws + WS_F2, 8, 8,
                                                       ws + WS_BPACK);
  k_attn<1><<<NNODE / 16, 128, 0, stream>>>(adj, ws + WS_F1, ws + WS_BPACK, ws + WS_MAX0,
                                            bias1, ws + WS_H, out);
  // Layer 2
  k_prep2<<<NNODE / 128, 128, 0, stream>>>(ws + WS_H, Wo, o1w, o1b, o2w, o2b, ws);
  k_rmax<<<1, 256, 0, stream>>>(ws + WS_F2B, NNODE, ws + WS_MAX1);
  k_bpack<<<(NNODE / 4) * 128 / 256, 256, 0, stream>>>(ws + WS_FTS2, ws + WS_F2B, 1, 1,
                                                       ws + WS_BPACK);
  k_attn<2><<<NNODE / 16, 128, 0, stream>>>(adj, ws + WS_F1B, ws + WS_BPACK, ws + WS_MAX1,
                                            biaso, nullptr, out);
}